// Tree_Net_5385888989359
// MI455X (gfx1250) — compile-verified
//
#include <hip/hip_runtime.h>
#include <math.h>

// Tree-Net for MI455X (gfx1250, wave32, WMMA).
// One block (8 wave32) per batch; the whole 127x512 f32 node matrix is
// LDS-resident (320KB/WGP on CDNA5). Circular correlation is two 16x512x16
// f32 GEMMs via a Toeplitz A operand (V_WMMA_F32_16X16X4_F32); operand
// streams are linearized into LDS so the WMMA loop runs on aligned
// ds_load_b64 with immediate offsets. Final 127x512 @ 512x511 linear layer is
// WMMA f32 with lin_w N-panels staged in LDS via b128 copies.

typedef __attribute__((ext_vector_type(2))) float v2f;
typedef __attribute__((ext_vector_type(8))) float v8f;

#define BATCH   512
#define NLEAF   64
#define NSTEP   63
#define NNODE   127
#define DIM     512
#define NCAT    511
#define ROWPAD  516              // 512 + 4 : LDS row stride (bank-conflict pad)
#define EPSN    1e-6f

#define VEC_FLOATS   (128 * ROWPAD)  // 128 rows (row 127 = zero pad row)
#define AUX_FLOATS   (16 * ROWPAD)   // lin_w panel (16x516) / reduction buffer
#define EXTA_SZ      536             // a[(i-16)%512], i in [0,528]
#define EXTA2_SZ     536             // a[(i-15)%512], i in [0,527]  (odd-m copy)
#define EXTB_SZ      1024            // b[i%512],      i in [0,1023]
#define OFF_AUX      (VEC_FLOATS)
#define OFF_EXTA     (OFF_AUX + AUX_FLOATS)
#define OFF_EXTA2    (OFF_EXTA + EXTA_SZ)
#define OFF_EXTB     (OFF_EXTA2 + EXTA2_SZ)
#define OFF_SS       (OFF_EXTB + EXTB_SZ)
#define SMEM_FLOATS  (OFF_SS + 16)

__global__ void __launch_bounds__(256, 1)
tree_net_kernel(const int* __restrict__ lci,       // (B,64,2) node_idx, vocab_id
                const int* __restrict__ cmask,     // (B,64)
                const int* __restrict__ cinfo,     // (B,63,4) num,parent,left,right
                const float* __restrict__ emb,     // (VOCAB,512)
                const float* __restrict__ lin_w,   // (511,512)
                const float* __restrict__ lin_b,   // (511)
                float* __restrict__ out)           // (B,127,511)
{
    extern __shared__ float smem[];
    float* VEC   = smem;
    float* AUX   = smem + OFF_AUX;
    float* EXTA  = smem + OFF_EXTA;
    float* EXTA2 = smem + OFF_EXTA2;
    float* EXTB  = smem + OFF_EXTB;
    float* SS    = smem + OFF_SS;

    const int b    = blockIdx.x;
    const int tid  = threadIdx.x;
    const int lane = tid & 31;
    const int wv   = tid >> 5;

    // Warm L2 with lin_w (1MB, L2-resident for all 512 blocks); emits
    // global_prefetch_b8.
    {
        const char* w8 = (const char*)lin_w;
        for (int i = tid * 16; i < (NCAT * DIM); i += 256 * 16)
            __builtin_prefetch(w8 + (size_t)i * 4, 0, 1);
    }

    // ---------------- Phase A: leaf gather + L2 normalize -------------------
    for (int ii = 0; ii < 8; ++ii) {
        const int i   = wv * 8 + ii;
        const int vid = lci[(b * NLEAF + i) * 2 + 1];
        const int msk = cmask[b * NLEAF + i];
        const float* erow = emb + (size_t)vid * DIM;
        float4 v[4];
        float ssl = 0.0f;
        #pragma unroll
        for (int c = 0; c < 4; ++c) {
            const int col = c * 128 + lane * 4;
            v[c] = *(const float4*)(erow + col);
            ssl += v[c].x * v[c].x + v[c].y * v[c].y + v[c].z * v[c].z + v[c].w * v[c].w;
        }
        #pragma unroll
        for (int off = 16; off > 0; off >>= 1) ssl += __shfl_xor(ssl, off, 32);
        const float scale = msk ? (1.0f / (sqrtf(ssl) + EPSN)) : 0.0f;
        float* row = VEC + i * ROWPAD;
        #pragma unroll
        for (int c = 0; c < 4; ++c) {
            const int col = c * 128 + lane * 4;
            *(float4*)(row + col) =
                make_float4(v[c].x * scale, v[c].y * scale, v[c].z * scale, v[c].w * scale);
        }
    }
    for (int k = tid; k < DIM; k += 256) VEC[127 * ROWPAD + k] = 0.0f;

    // ---------------- Phase B: 63 sequential composition steps --------------
    // c[m + 16n + 256h] = sum_j a[(j-m)%512] * b[(j + 16n + 256h)%512]
    // wave -> (h = wv>>2, K-quarter q = wv&3). Operands read from linearized
    // EXTA/EXTA2/EXTB streams: per-lane base pointers are loop-invariant and
    // 8-byte aligned -> ds_load_b64 with immediate offsets.
    const int h  = wv >> 2;
    const int q  = wv & 3;
    const int m  = lane & 15;                   // A row (both half-waves)
    const int ko = (lane >> 4) * 2;             // K sub-offset 0 / 2
    const int nb = 16 * (lane & 15) + 256 * h;  // B column shift
    const int mb = (lane >> 4) * 8;             // D-layout M base

    const int j0    = q * 128;
    const int abase = j0 + ko - m + 16;
    const float* pa = (m & 1) ? (EXTA2 + (abase - 1)) : (EXTA + abase);
    const float* pb = EXTB + (j0 + ko + nb);

    for (int t = 0; t < NSTEP; ++t) {
        __syncthreads();                        // prior step's writes visible
        const int* ci = cinfo + (b * NSTEP + t) * 4;
        const int num = ci[0], p = ci[1], l = ci[2], r = ci[3];
        const float* arow = VEC + l * ROWPAD;
        const float* brow = VEC + r * ROWPAD;

        if (num == 2) {
            // linearize operand streams (resolve circular wrap once)
            for (int i = tid; i < 529; i += 256)  EXTA[i]  = arow[(i + 496) & 511];
            for (int i = tid; i < 528; i += 256)  EXTA2[i] = arow[(i + 497) & 511];
            for (int i = tid; i < 1024; i += 256) EXTB[i]  = brow[i & 511];
            if (tid == 0) SS[0] = 0.0f;
            __syncthreads();

            v8f acc = {};
            #pragma unroll
            for (int jj = 0; jj < 128; jj += 4) {
                const v2f av = *(const v2f*)(pa + jj);
                const v2f bv = *(const v2f*)(pb + jj);
                acc = __builtin_amdgcn_wmma_f32_16x16x4_f32(
                        false, av, false, bv, (short)0, acc, false, false);
            }
            // scatter partial tile: element k = 256h + (mb+rr) + 16*(lane&15)
            #pragma unroll
            for (int rr = 0; rr < 8; ++rr) {
                const int k = 256 * h + mb + rr + 16 * (lane & 15);
                AUX[q * 512 + k] = acc[rr];
            }
            __syncthreads();
            // reduce 4 K-partials, accumulate squared norm
            const int k0 = tid, k1 = tid + 256;
            const float c0 = AUX[k0] + AUX[512 + k0] + AUX[1024 + k0] + AUX[1536 + k0];
            const float c1 = AUX[k1] + AUX[512 + k1] + AUX[1024 + k1] + AUX[1536 + k1];
            atomicAdd(&SS[0], c0 * c0 + c1 * c1);   // ds_add_f32
            __syncthreads();
            const float scale = 1.0f / (sqrtf(SS[0]) + EPSN);
            float* prow = VEC + p * ROWPAD;
            prow[tid]       = c0 * scale;
            prow[tid + 256] = c1 * scale;
        } else if (num > 0) {
            float* prow = VEC + p * ROWPAD;
            prow[tid]       = arow[tid];
            prow[tid + 256] = arow[tid + 256];
        }
    }

    // ---------------- Phase C: out = VEC @ lin_w^T + lin_b ------------------
    // M = 128 (row 127 zero-padded), N = 512 (col 511 padded), K = 512.
    // wave wv owns M-tile wv; 32 N-tiles; lin_w panel staged in LDS (b128).
    float* panel = AUX;                          // [16][ROWPAD], n-major
    float* outb  = out + (size_t)b * NNODE * NCAT;
    const float* ap = VEC + (wv * 16 + m) * ROWPAD + ko;

    for (int nt = 0; nt < 32; ++nt) {
        __syncthreads();                         // previous panel consumed
        {   // cooperative panel load: thread -> (n = tid>>4, 32 consecutive k)
            const int n   = tid >> 4;
            const int col = nt * 16 + n;
            const int kb  = (tid & 15) * 32;
            float4* dst = (float4*)(panel + n * ROWPAD + kb);
            if (col < NCAT) {
                const float4* src = (const float4*)(lin_w + (size_t)col * DIM + kb);
                #pragma unroll
                for (int j = 0; j < 8; ++j) dst[j] = src[j];
            } else {
                #pragma unroll
                for (int j = 0; j < 8; ++j) dst[j] = make_float4(0.f, 0.f, 0.f, 0.f);
            }
        }
        __syncthreads();

        v8f acc = {};
        const float* bp = panel + (lane & 15) * ROWPAD + ko;
        #pragma unroll 8
        for (int j = 0; j < DIM; j += 4) {
            const v2f av = *(const v2f*)(ap + j);
            const v2f bv = *(const v2f*)(bp + j);
            acc = __builtin_amdgcn_wmma_f32_16x16x4_f32(
                    false, av, false, bv, (short)0, acc, false, false);
        }

        const int col   = nt * 16 + (lane & 15);
        const float bias = (col < NCAT) ? lin_b[col] : 0.0f;
        const int rowb  = wv * 16 + mb;
        #pragma unroll
        for (int rr = 0; rr < 8; ++rr) {
            const int row = rowb + rr;
            if (row < NNODE && col < NCAT)
                outb[(size_t)row * NCAT + col] = acc[rr] + bias;
        }
    }
}

extern "C" void kernel_launch(void* const* d_in, const int* in_sizes, int n_in,
                              void* d_out, int out_size, void* d_ws, size_t ws_size,
                              hipStream_t stream) {
    const int*   lci   = (const int*)d_in[0];
    const int*   cmask = (const int*)d_in[1];
    const int*   cinfo = (const int*)d_in[2];
    const float* emb   = (const float*)d_in[3];
    const float* lin_w = (const float*)d_in[4];
    const float* lin_b = (const float*)d_in[5];
    float* out = (float*)d_out;

    const size_t shmem = (size_t)SMEM_FLOATS * sizeof(float);
    tree_net_kernel<<<dim3(BATCH), dim3(256), shmem, stream>>>(
        lci, cmask, cinfo, emb, lin_w, lin_b, out);
}